// DocLSTM_34832184771191
// MI455X (gfx1250) — compile-verified
//
#include <hip/hip_runtime.h>
#include <float.h>
#include <math.h>

#define SNT 512                 // S sentences
#define DD  512                 // D dim
#define SD  (512 * 512)
#define NF  10

typedef float v2f __attribute__((ext_vector_type(2)));
typedef float v8f __attribute__((ext_vector_type(8)));

// ---------------------------------------------------------------------------
// WMMA helper: exact fp32 matmul on the matrix pipe (V_WMMA_F32_16X16X4_F32)
// ---------------------------------------------------------------------------
__device__ __forceinline__ v8f wmma_f32_16x16x4(v2f a, v2f b, v8f c) {
  // 8 args: (neg_a, A, neg_b, B, c_mod, C, reuse_a, reuse_b)
  return __builtin_amdgcn_wmma_f32_16x16x4_f32(false, a, false, b, (short)0, c,
                                               false, false);
}

// One wave computes a 16x64 tile of C = op(A) * op(B) (+bias) ; N fixed = 512.
// TA: A is stored transposed (A[k*lda + m]); TB: B stored transposed (B[n*ldb + k]).
// Rows m >= validCnt are written as 0 (post-bias mask, matches jnp.where(mask,..,0)).
template <int TA, int TB>
__device__ __forceinline__ void gemm_tile(const float* __restrict__ A, int lda,
                                          const float* __restrict__ B, int ldb,
                                          float* __restrict__ C, int ldc,
                                          const float* __restrict__ bias, int K,
                                          int validCnt) {
  const int lane = threadIdx.x & 31;   // wave32
  const int wave = threadIdx.x >> 5;
  const int tile = blockIdx.x * 8 + wave;     // 8 waves / 256-thread block
  const int m0   = (tile >> 3) * 16;          // 512/64 = 8 n-tiles
  const int n0   = (tile & 7) * 64;
  const int l    = lane & 15;
  const int half = lane >> 4;
  const int arow = m0 + l;

  v8f acc[4];
#pragma unroll
  for (int t = 0; t < 4; ++t)
    acc[t] = (v8f){0.f, 0.f, 0.f, 0.f, 0.f, 0.f, 0.f, 0.f};

  for (int k = 0; k < K; k += 4) {
    const int ka = k + half * 2;
    v2f a;
    if (TA) {
      a.x = A[(size_t)ka * lda + arow];
      a.y = A[(size_t)(ka + 1) * lda + arow];
    } else {
      const float* ap = A + (size_t)arow * lda + ka;
      a.x = ap[0];
      a.y = ap[1];
    }
#pragma unroll
    for (int t = 0; t < 4; ++t) {
      const int col = n0 + t * 16 + l;
      v2f b;
      if (TB) {
        const float* bp = B + (size_t)col * ldb + ka;
        b.x = bp[0];
        b.y = bp[1];
      } else {
        b.x = B[(size_t)ka * ldb + col];
        b.y = B[(size_t)(ka + 1) * ldb + col];
      }
      acc[t] = wmma_f32_16x16x4(a, b, acc[t]);
    }
  }

#pragma unroll
  for (int v = 0; v < 8; ++v) {
    const int m = m0 + v + half * 8;
    const bool dead = (m >= validCnt);
#pragma unroll
    for (int t = 0; t < 4; ++t) {
      const int n = n0 + t * 16 + l;
      float val = acc[t][v];
      if (bias) val += bias[n];
      C[(size_t)m * ldc + n] = dead ? 0.0f : val;
    }
  }
}

// ---------------------------------------------------------------------------
// Stage 1: u = r @ W_sim   (u[b] = sum_a r[a] * W[a*D+b])
// ---------------------------------------------------------------------------
__global__ void k_u(const float* __restrict__ r, const float* __restrict__ W,
                    float* __restrict__ u) {
  const int col = blockIdx.x * 256 + threadIdx.x;
  float acc = 0.f;
  for (int a = 0; a < DD; ++a) acc = fmaf(r[a], W[a * DD + col], acc);
  u[col] = acc;
}

// Stage 2: head_sim[n] = sigmoid(dot(u, sent[n]) + b_sim)
__global__ void k_headsim(const float* __restrict__ u,
                          const float* __restrict__ sent,
                          const float* __restrict__ bsim,
                          float* __restrict__ hs) {
  __shared__ float red[64];
  const int n = blockIdx.x;
  float acc = 0.f;
  for (int d = threadIdx.x; d < DD; d += 64)
    acc = fmaf(u[d], sent[(size_t)n * DD + d], acc);
  red[threadIdx.x] = acc;
  __syncthreads();
  for (int s = 32; s > 0; s >>= 1) {
    if (threadIdx.x < s) red[threadIdx.x] += red[threadIdx.x + s];
    __syncthreads();
  }
  if (threadIdx.x == 0) {
    float x = red[0] + bsim[0];
    hs[n] = 1.0f / (1.0f + expf(-x));
  }
}

// Stage 3: softmax(head_sim) -> prob ; stable compaction indices hi/lo + counts
__global__ void k_softmax_compact(const float* __restrict__ hs,
                                  float* __restrict__ prob,
                                  int* __restrict__ cnts, int* __restrict__ hi,
                                  int* __restrict__ lo) {
  __shared__ float red[SNT];
  const int t = threadIdx.x;
  const float v = hs[t];
  red[t] = v;
  __syncthreads();
  for (int s = 256; s > 0; s >>= 1) {
    if (t < s) red[t] = fmaxf(red[t], red[t + s]);
    __syncthreads();
  }
  const float mx = red[0];
  __syncthreads();
  const float e = expf(v - mx);
  red[t] = e;
  __syncthreads();
  for (int s = 256; s > 0; s >>= 1) {
    if (t < s) red[t] += red[t + s];
    __syncthreads();
  }
  prob[t] = e / red[0];
  __syncthreads();
  if (t == 0) {
    int nh = 0;
    for (int n = 0; n < SNT; ++n)
      if (hs[n] > 0.5f) hi[nh++] = n;
    cnts[0] = nh;
    int nl = 0;
    for (int n = 0; n < SNT; ++n)
      if (!(hs[n] > 0.5f)) lo[nl++] = n;
    cnts[1] = nl;
  }
}

// Stage 4: materialize high_x (b=0) / low_x (b=1): compacted prob-weighted rows
__global__ void k_buildx(const float* __restrict__ sent,
                         const float* __restrict__ prob,
                         const int* __restrict__ cnts, const int* __restrict__ hi,
                         const int* __restrict__ lo, float* __restrict__ x) {
  const int s = blockIdx.x;
  const int b = blockIdx.y;
  const int cnt = cnts[b];
  const int src = (s < cnt) ? (b == 0 ? hi[s] : lo[s]) : -1;
  const float p = (src >= 0) ? prob[src] : 0.f;
  float* xr = x + (size_t)b * SD + (size_t)s * DD;
  if (src >= 0) {
    const float* sr = sent + (size_t)src * DD;
    for (int d = threadIdx.x; d < DD; d += 256) xr[d] = p * sr[d];
  } else {
    for (int d = threadIdx.x; d < DD; d += 256) xr[d] = 0.0f;
  }
}

// Stage 5: conv (full padding) + global max pool + bias, per (branch, ws, filter)
// float4 inner loop: b128 global + ds_load_b128 instead of 4x b32.
__global__ void k_conv(const float* __restrict__ x, const float* cw1,
                       const float* cw2, const float* cw3, const float* cw4,
                       const float* cw5, const float* cw6, const float* cw7,
                       const float* __restrict__ convb, float* __restrict__ out) {
  __shared__ __align__(16) float kw[7 * DD];
  __shared__ float red[256];
  const int id = blockIdx.x;  // 0..139
  const int b = id / 70;
  const int rem = id % 70;
  const int wi = rem / NF;
  const int f = rem % NF;
  const int ws = wi + 1;
  const float* cw;
  switch (wi) {
    case 0: cw = cw1; break;
    case 1: cw = cw2; break;
    case 2: cw = cw3; break;
    case 3: cw = cw4; break;
    case 4: cw = cw5; break;
    case 5: cw = cw6; break;
    default: cw = cw7; break;
  }
  const int klen = ws * DD;
  for (int i = threadIdx.x; i < klen; i += 256) kw[i] = cw[f * klen + i];
  __syncthreads();
  const float* xb = x + (size_t)b * SD;
  float best = -FLT_MAX;
  const int T = SNT + ws - 1;
  for (int t = threadIdx.x; t < T; t += 256) {
    float acc = 0.f;
    for (int j = 0; j < ws; ++j) {
      const int r = t + j - (ws - 1);
      if (r < 0 || r >= SNT) continue;
      const float4* xr4 = (const float4*)(xb + (size_t)r * DD);
      const float4* kr4 = (const float4*)(kw + j * DD);
      float a0 = 0.f, a1 = 0.f, a2 = 0.f, a3 = 0.f;
      for (int d = 0; d < DD / 4; ++d) {
        const float4 xv = xr4[d];
        const float4 kv = kr4[d];
        a0 = fmaf(xv.x, kv.x, a0);
        a1 = fmaf(xv.y, kv.y, a1);
        a2 = fmaf(xv.z, kv.z, a2);
        a3 = fmaf(xv.w, kv.w, a3);
      }
      acc += (a0 + a1) + (a2 + a3);
    }
    best = fmaxf(best, acc);
  }
  red[threadIdx.x] = best;
  __syncthreads();
  for (int s = 128; s > 0; s >>= 1) {
    if (threadIdx.x < s)
      red[threadIdx.x] = fmaxf(red[threadIdx.x], red[threadIdx.x + s]);
    __syncthreads();
  }
  // concat order: [high_feat, low_feat, high_pooled(=low_sim_conv_vec), low_pooled]
  if (threadIdx.x == 0)
    out[1024 + b * 70 + wi * NF + f] = red[0] + convb[wi * NF + f];
}

// Stage 6: 12x (512^3) QKV GEMMs on WMMA.  qkv[((b*2+h)*3+m)] = x[b]@W[h] + bias[h]
__global__ void k_gemm_qkv(const float* __restrict__ x, const float* Wq,
                           const float* Wk, const float* Wv, const float* bq,
                           const float* bk, const float* bv,
                           float* __restrict__ qkv) {
  const int y = blockIdx.y;
  const int b = y / 6, r = y % 6, h = r / 3, m = r % 3;
  const float* W = (m == 0 ? Wq : (m == 1 ? Wk : Wv)) + (size_t)h * DD * DD;
  const float* bias = (m == 0 ? bq : (m == 1 ? bk : bv)) + h * DD;
  const float* A = x + (size_t)b * SD;
  float* C = qkv + (size_t)((b * 2 + h) * 3 + m) * SD;
  gemm_tile<0, 0>(A, DD, W, DD, C, DD, bias, DD, 1 << 30);
}

// Stage 7: simT = K @ Q^T   (so softmax over query axis is a contiguous row op)
__global__ void k_gemm_scores(const float* __restrict__ qkv,
                              float* __restrict__ w) {
  const int bh = blockIdx.y;  // 0..3
  const float* Km = qkv + (size_t)(bh * 3 + 1) * SD;
  const float* Qm = qkv + (size_t)(bh * 3 + 0) * SD;
  float* C = w + (size_t)bh * SD;
  gemm_tile<0, 1>(Km, DD, Qm, DD, C, SNT, nullptr, DD, 1 << 30);
}

// Stage 8: per column j of sim (= row j of simT): indicator, scale, query-row
// mask, softmax over dim 0, then key-column mask. In place on w.
__global__ void k_attn_softmax(float* __restrict__ w,
                               const int* __restrict__ cnts) {
  __shared__ float red[256];
  const int bh = blockIdx.y;
  const int b = bh >> 1;  // b=0: indicator 1 (high), b=1: indicator 0 (low)
  const int j = blockIdx.x;
  const int t = threadIdx.x;
  const int cnt = cnts[b];
  const float inv = 0.044194173824159216f;  // 1/sqrt(512)
  float* row = w + (size_t)bh * SD + (size_t)j * SNT;
  float v0 = row[t], v1 = row[t + 256];
  if (b == 1) { v0 = 1.0f - v0; v1 = 1.0f - v1; }
  v0 *= inv;
  v1 *= inv;
  if (t >= cnt) v0 = -FLT_MAX;
  if (t + 256 >= cnt) v1 = -FLT_MAX;
  red[t] = fmaxf(v0, v1);
  __syncthreads();
  for (int s = 128; s > 0; s >>= 1) {
    if (t < s) red[t] = fmaxf(red[t], red[t + s]);
    __syncthreads();
  }
  const float mx = red[0];
  __syncthreads();
  const float e0 = expf(v0 - mx), e1 = expf(v1 - mx);
  red[t] = e0 + e1;
  __syncthreads();
  for (int s = 128; s > 0; s >>= 1) {
    if (t < s) red[t] += red[t + s];
    __syncthreads();
  }
  const float denom = red[0];
  const float km = (j < cnt) ? 1.0f : 0.0f;
  row[t] = km * e0 / denom;
  row[t + 256] = km * e1 / denom;
}

// Stage 9: head out -> hc columns:  out = wT^T @ V  (TN GEMM)
__global__ void k_gemm_wv(const float* __restrict__ w,
                          const float* __restrict__ qkv,
                          float* __restrict__ hc) {
  const int bh = blockIdx.y;
  const int b = bh >> 1, h = bh & 1;
  const float* A = w + (size_t)bh * SD;                 // wT [S x S]
  const float* V = qkv + (size_t)(bh * 3 + 2) * SD;     // V  [S x D]
  float* C = hc + (size_t)b * SNT * 1024 + (size_t)h * DD;
  gemm_tile<1, 0>(A, SNT, V, DD, C, 1024, nullptr, SNT, 1 << 30);
}

// Stage 10: mo = hc @ W_cat + b_cat, padded rows zeroed
__global__ void k_gemm_cat(const float* __restrict__ hc,
                           const float* __restrict__ Wcat,
                           const float* __restrict__ bcat,
                           const int* __restrict__ cnts, float* __restrict__ mo) {
  const int b = blockIdx.y;
  const int cnt = cnts[b];
  const float* A = hc + (size_t)b * SNT * 1024;
  float* C = mo + (size_t)b * SD;
  gemm_tile<0, 0>(A, 1024, Wcat, DD, C, DD, bcat, 1024, cnt);
}

// Stage 11: stream W_feat (512 MB) ONCE, both branches per pass. HBM-bound:
// 512MB / 23.3 TB/s ~= 22us floor. One b128 load per thread covers 2 rows per
// block step (256 lanes x 16B = 4KB = 2 full rows). Threads 0..127 accumulate
// even rows, 128..255 odd rows -> extra parity slot in the partials.
// Deterministic (no atomics).
__global__ void k_feat_partial(const float* __restrict__ mo,
                               const float* __restrict__ Wf,
                               float* __restrict__ part) {
  const int c = blockIdx.x;        // 256 chunks of 1024 rows
  const int t = threadIdx.x;       // 0..255
  const int p = t >> 7;            // row parity this thread covers
  const int col = (t * 4) & 511;   // 4 owned columns
  const float* moh = mo;
  const float* mol = mo + SD;
  float ah0 = 0.f, ah1 = 0.f, ah2 = 0.f, ah3 = 0.f;
  float al0 = 0.f, al1 = 0.f, al2 = 0.f, al3 = 0.f;
  const int i0 = c * 1024;
#pragma unroll 4
  for (int i = i0; i < i0 + 1024; i += 2) {
    const float4 w = *(const float4*)(Wf + (size_t)i * DD + t * 4);
    __builtin_prefetch(Wf + (size_t)(i + 16) * DD + t * 4, 0, 0);  // 32KB ahead
    const int r = i + p;
    const float mh = moh[r];
    const float ml = mol[r];
    ah0 = fmaf(mh, w.x, ah0);
    ah1 = fmaf(mh, w.y, ah1);
    ah2 = fmaf(mh, w.z, ah2);
    ah3 = fmaf(mh, w.w, ah3);
    al0 = fmaf(ml, w.x, al0);
    al1 = fmaf(ml, w.y, al1);
    al2 = fmaf(ml, w.z, al2);
    al3 = fmaf(ml, w.w, al3);
  }
  // part layout: [c][parity][branch][512]
  float* pp = part + (size_t)c * 2048 + (size_t)p * 1024;
  pp[col + 0] = ah0;
  pp[col + 1] = ah1;
  pp[col + 2] = ah2;
  pp[col + 3] = ah3;
  pp[512 + col + 0] = al0;
  pp[512 + col + 1] = al1;
  pp[512 + col + 2] = al2;
  pp[512 + col + 3] = al3;
}

__global__ void k_feat_reduce(const float* __restrict__ part,
                              const float* __restrict__ bfeat,
                              float* __restrict__ out) {
  const int q = blockIdx.x * 256 + threadIdx.x;  // 0..1023 = branch*512 + n
  float s = 0.f;
  for (int j = 0; j < 512; ++j) s += part[(size_t)j * 1024 + q];
  out[q] = s + bfeat[q & 511];
}

// ---------------------------------------------------------------------------
extern "C" void kernel_launch(void* const* d_in, const int* in_sizes, int n_in,
                              void* d_out, int out_size, void* d_ws,
                              size_t ws_size, hipStream_t stream) {
  const float* sent  = (const float*)d_in[0];
  const float* rsent = (const float*)d_in[1];
  const float* Wsim  = (const float*)d_in[2];
  const float* bsim  = (const float*)d_in[3];
  const float* Wq    = (const float*)d_in[4];
  const float* Wk    = (const float*)d_in[5];
  const float* Wv    = (const float*)d_in[6];
  const float* bq    = (const float*)d_in[7];
  const float* bk    = (const float*)d_in[8];
  const float* bv    = (const float*)d_in[9];
  const float* Wcat  = (const float*)d_in[10];
  const float* bcat  = (const float*)d_in[11];
  const float* Wfeat = (const float*)d_in[12];
  const float* bfeat = (const float*)d_in[13];
  const float* convb = (const float*)d_in[14];
  const float* cw1 = (const float*)d_in[15];
  const float* cw2 = (const float*)d_in[16];
  const float* cw3 = (const float*)d_in[17];
  const float* cw4 = (const float*)d_in[18];
  const float* cw5 = (const float*)d_in[19];
  const float* cw6 = (const float*)d_in[20];
  const float* cw7 = (const float*)d_in[21];
  float* out = (float*)d_out;

  // workspace layout (floats); total ~27 MB
  float* ws = (float*)d_ws;
  float* u    = ws + 0;
  float* hs   = ws + 512;
  float* prob = ws + 1024;
  int*   cnts = (int*)(ws + 1536);
  int*   hi   = (int*)(ws + 1544);
  int*   lo   = (int*)(ws + 2056);
  float* x    = ws + 4096;                 // [2][S][D]
  float* qkv  = x + 2 * (size_t)SD;        // [2][2][3][S][D]
  float* wbuf = qkv + 12 * (size_t)SD;     // [2][2][S][S]  (simT -> w)
  float* hc   = wbuf + 4 * (size_t)SD;     // [2][S][1024]
  float* mo   = hc + 4 * (size_t)SD;       // [2][S][D]
  float* part = mo + 2 * (size_t)SD;       // [256][2][2][512]

  k_u<<<2, 256, 0, stream>>>(rsent, Wsim, u);
  k_headsim<<<SNT, 64, 0, stream>>>(u, sent, bsim, hs);
  k_softmax_compact<<<1, 512, 0, stream>>>(hs, prob, cnts, hi, lo);
  k_buildx<<<dim3(SNT, 2), 256, 0, stream>>>(sent, prob, cnts, hi, lo, x);
  k_conv<<<140, 256, 0, stream>>>(x, cw1, cw2, cw3, cw4, cw5, cw6, cw7, convb,
                                  out);
  k_gemm_qkv<<<dim3(32, 12), 256, 0, stream>>>(x, Wq, Wk, Wv, bq, bk, bv, qkv);
  k_gemm_scores<<<dim3(32, 4), 256, 0, stream>>>(qkv, wbuf);
  k_attn_softmax<<<dim3(SNT, 4), 256, 0, stream>>>(wbuf, cnts);
  k_gemm_wv<<<dim3(32, 4), 256, 0, stream>>>(wbuf, qkv, hc);
  k_gemm_cat<<<dim3(32, 2), 256, 0, stream>>>(hc, Wcat, bcat, cnts, mo);
  k_feat_partial<<<256, 256, 0, stream>>>(mo, Wfeat, part);
  k_feat_reduce<<<4, 256, 0, stream>>>(part, bfeat, out);
}